// AlignedGloveLayer_55387898249622
// MI455X (gfx1250) — compile-verified
//
#include <hip/hip_runtime.h>
#include <hip/hip_bf16.h>

// ---------------------------------------------------------------------------
// mismatch-loss: out = (1/K) * count( argmin_j ||mapped[idx[k]] - target[j]||
//                                     != idx[k] )
// argmin_j d2 == argmax_j ( dot(a_k, t_j) - 0.5*||t_j||^2 )
// fp32 WMMA (v_wmma_f32_16x16x4_f32) for exact-precision GEMM + fused argmax.
// ---------------------------------------------------------------------------

typedef float v2f __attribute__((ext_vector_type(2)));
typedef float v8f __attribute__((ext_vector_type(8)));

#define NXc 30000
#define NYc 30000
#define Dc  256
#define Kc  4096

#define NWAVES 5     // waves per block
#define NTILE  5     // consecutive 16-col tiles per wave per iteration
#define ROWT   2     // 16-row tiles per block (32 rows/block)
#define NCOLT  (NYc / 16)                  // 1875 column tiles total
#define NITER  (NCOLT / (NWAVES * NTILE))  // 75, exact

// ---------------------------------------------------------------------------
// Kernel 1: b2h[j] = 0.5 * sum_d target[j,d]^2 ; also zero mismatch counter.
// One wave32 per row, float4 coalesced loads, shuffle reduction.
// ---------------------------------------------------------------------------
__global__ void prep_kernel(const float* __restrict__ target,
                            float* __restrict__ b2h,
                            int* __restrict__ count) {
  if (blockIdx.x == 0 && threadIdx.x == 0) *count = 0;
  const int wave = threadIdx.x >> 5;
  const int lane = threadIdx.x & 31;
  const int row  = blockIdx.x * 8 + wave;
  if (row >= NYc) return;
  const float4* t4 = (const float4*)(target + (size_t)row * Dc);
  float s = 0.f;
#pragma unroll
  for (int it = 0; it < Dc / (4 * 32); ++it) {   // 2 iterations
    float4 v = t4[lane + it * 32];
    s += v.x * v.x + v.y * v.y + v.z * v.z + v.w * v.w;
  }
#pragma unroll
  for (int m = 16; m >= 1; m >>= 1) s += __shfl_xor(s, m, 32);
  if (lane == 0) b2h[row] = 0.5f * s;
}

// ---------------------------------------------------------------------------
// Kernel 2: fused GEMM + running argmax + mismatch count.
// Block = 160 threads (5 waves), 32 query rows per block (2 row tiles).
// Wave w scans column tiles [(it*5 + w)*5 .. +5) for it in [0,75).
// A fragments (16x4 f32): lane m = lane&15, K = kk + 2*(lane>>4) + {0,1}
// B fragments (4x16 f32): lane n = lane&15, K = kk + 2*(lane>>4) + {0,1}
// C/D (16x16 f32): slot r -> M = r + 8*(lane>>4), N = lane&15
// ---------------------------------------------------------------------------
__global__ __launch_bounds__(NWAVES * 32)
void nn_kernel(const float* __restrict__ mapped,
               const float* __restrict__ target,
               const int* __restrict__ indexes,
               const float* __restrict__ b2h,
               int* __restrict__ count) {
  __shared__ float ldsA[ROWT * 16 * Dc];        // 32 KB gathered A tile
  __shared__ float sBest[NWAVES][ROWT * 16];
  __shared__ int   sIdx[NWAVES][ROWT * 16];

  const int tid = threadIdx.x;
  const int k0  = blockIdx.x * (ROWT * 16);     // first query row of block

  // Gather 32 rows of mapped[indexes[...]] into LDS (float4 granularity).
  for (int l = tid; l < ROWT * 16 * (Dc / 4); l += NWAVES * 32) {
    const int row = l / (Dc / 4);
    const int c4  = l % (Dc / 4);
    const int src = indexes[k0 + row];
    ((float4*)ldsA)[row * (Dc / 4) + c4] =
        ((const float4*)(mapped + (size_t)src * Dc))[c4];
  }
  __syncthreads();

  const int w    = tid >> 5;
  const int lane = tid & 31;
  const int g    = lane >> 4;   // lane half-group
  const int r16  = lane & 15;   // A-row / B-col within tile
  const int ko   = 2 * g;       // K sub-offset of this half-group

  float best[ROWT][8];
  int   bidx[ROWT][8];
#pragma unroll
  for (int rt = 0; rt < ROWT; ++rt)
#pragma unroll
    for (int r = 0; r < 8; ++r) { best[rt][r] = -3.4e38f; bidx[rt][r] = 0x7fffffff; }

  const float* a0p = ldsA + (size_t)r16 * Dc;         // row tile 0
  const float* a1p = ldsA + (size_t)(16 + r16) * Dc;  // row tile 1
  const v8f vzero = {0.f, 0.f, 0.f, 0.f, 0.f, 0.f, 0.f, 0.f};

  for (int it = 0; it < NITER; ++it) {
    const int tbase = (it * NWAVES + w) * NTILE;      // 5 consecutive col tiles

    v8f acc[ROWT][NTILE];
#pragma unroll
    for (int rt = 0; rt < ROWT; ++rt)
#pragma unroll
      for (int u = 0; u < NTILE; ++u) acc[rt][u] = vzero;

    const float* bp[NTILE];
#pragma unroll
    for (int u = 0; u < NTILE; ++u)
      bp[u] = target + (size_t)((tbase + u) * 16 + r16) * Dc + ko;

    for (int kk = 0; kk < Dc; kk += 4) {
      const v2f a0 = *(const v2f*)(a0p + kk + ko);    // ds_load_b64
      const v2f a1 = *(const v2f*)(a1p + kk + ko);    // ds_load_b64
#pragma unroll
      for (int u = 0; u < NTILE; ++u) {
        const v2f b = *(const v2f*)(bp[u] + kk);      // global_load_b64 (L2 hit)
        acc[0][u] = __builtin_amdgcn_wmma_f32_16x16x4_f32(
            false, a0, false, b, (short)0, acc[0][u], false, false);
        acc[1][u] = __builtin_amdgcn_wmma_f32_16x16x4_f32(
            false, a1, false, b, (short)0, acc[1][u], false, false);
      }
    }

    // score = dot - 0.5*||t||^2 ; running argmax (strict > keeps lowest col)
#pragma unroll
    for (int u = 0; u < NTILE; ++u) {
      const int col = (tbase + u) * 16 + r16;
      const float bh = b2h[col];
#pragma unroll
      for (int rt = 0; rt < ROWT; ++rt)
#pragma unroll
        for (int r = 0; r < 8; ++r) {
          const float s = acc[rt][u][r] - bh;
          if (s > best[rt][r]) { best[rt][r] = s; bidx[rt][r] = col; }
        }
    }
  }

  // Reduce across the 16 lanes of each half-group (xor masks stay in-group).
#pragma unroll
  for (int rt = 0; rt < ROWT; ++rt)
#pragma unroll
    for (int r = 0; r < 8; ++r) {
      float b = best[rt][r];
      int   i = bidx[rt][r];
#pragma unroll
      for (int m = 8; m >= 1; m >>= 1) {
        const float ob = __shfl_xor(b, m, 32);
        const int   oi = __shfl_xor(i, m, 32);
        if (ob > b || (ob == b && oi < i)) { b = ob; i = oi; }
      }
      if (r16 == 0) {                       // lanes 0 and 16 hold group winners
        const int rowLocal = rt * 16 + g * 8 + r;   // M = r + 8g within tile rt
        sBest[w][rowLocal] = b;
        sIdx[w][rowLocal]  = i;
      }
    }
  __syncthreads();

  // Combine the 5 waves' partial argmaxes; count mismatches.
  if (tid < ROWT * 16) {
    float b = sBest[0][tid];
    int   i = sIdx[0][tid];
#pragma unroll
    for (int w2 = 1; w2 < NWAVES; ++w2) {
      const float ob = sBest[w2][tid];
      const int   oi = sIdx[w2][tid];
      if (ob > b || (ob == b && oi < i)) { b = ob; i = oi; }
    }
    const int mism = (i != indexes[k0 + tid]) ? 1 : 0;
    atomicAdd(count, mism);   // integer atomics -> deterministic
  }
}

// ---------------------------------------------------------------------------
// Kernel 3: scalar finalize.
// ---------------------------------------------------------------------------
__global__ void finalize_kernel(const int* __restrict__ count,
                                float* __restrict__ out) {
  if (blockIdx.x == 0 && threadIdx.x == 0)
    out[0] = (float)(*count) * (1.0f / (float)Kc);
}

extern "C" void kernel_launch(void* const* d_in, const int* in_sizes, int n_in,
                              void* d_out, int out_size, void* d_ws, size_t ws_size,
                              hipStream_t stream) {
  const float* mapped  = (const float*)d_in[0];   // [NX, D] fp32
  const float* target  = (const float*)d_in[1];   // [NY, D] fp32
  const int*   indexes = (const int*)d_in[2];     // [K] int32
  float* out = (float*)d_out;

  // Workspace: [0] int counter ; b2h at byte offset 256 (30000 floats).
  int*   count = (int*)d_ws;
  float* b2h   = (float*)((char*)d_ws + 256);

  prep_kernel<<<NYc / 8, 256, 0, stream>>>(target, b2h, count);
  nn_kernel<<<Kc / (ROWT * 16), NWAVES * 32, 0, stream>>>(mapped, target,
                                                          indexes, b2h, count);
  finalize_kernel<<<1, 32, 0, stream>>>(count, out);
}